// InfoNCE_5823975653903
// MI455X (gfx1250) — compile-verified
//
#include <hip/hip_runtime.h>
#include <math.h>

#define NS   512      // samples
#define XD   768      // x/y feature dim
#define HID  300      // hidden
#define HP   304      // hidden padded to 16
#define LDW  1536     // W1 row stride (x|y concatenated)

typedef __attribute__((ext_vector_type(2))) float v2f;
typedef __attribute__((ext_vector_type(8))) float v8f;

// ---------------------------------------------------------------------------
// Kernel 1: phx[n,h] = sum_d X[n,d]*W1[h,d] + b1[h]   (z==0)
//           phy[n,h] = sum_d Y[n,d]*W1[h,768+d]        (z==1)
// One wave (32 lanes) computes one 16x16 output tile via V_WMMA_F32_16X16X4_F32,
// looping K over 768 in steps of 4 (192 WMMAs). Pad columns h>=300 written 0.
// ---------------------------------------------------------------------------
__global__ void hid_gemm_wmma(const float* __restrict__ X,
                              const float* __restrict__ Y,
                              const float* __restrict__ W1,
                              const float* __restrict__ b1,
                              float* __restrict__ phx,
                              float* __restrict__ phy) {
  const int lane = threadIdx.x;           // blockDim.x == 32, one wave
  const int n0 = blockIdx.x * 16;         // sample tile
  const int h0 = blockIdx.y * 16;         // hidden tile
  const int z  = blockIdx.z;              // 0 = x-path, 1 = y-path
  const float* A   = z ? Y : X;
  float*       Dst = z ? phy : phx;
  const int woff = z * XD;

  const int mr   = lane & 15;             // A-row / B-col within tile
  const int koff = (lane < 16) ? 0 : 2;   // f32 WMMA: lanes 0-15 hold K{0,1}, 16-31 hold K{2,3}

  const float* pa = A + (size_t)(n0 + mr) * XD + koff;

  const int  hb     = h0 + mr;
  const bool bvalid = (hb < HID);
  // clamp pointer to a valid row; mask contribution arithmetically
  const float* pb   = W1 + (size_t)(bvalid ? hb : 0) * LDW + woff + koff;
  const float  bmsk = bvalid ? 1.0f : 0.0f;

  v8f acc = {};
  #pragma unroll 4
  for (int k0 = 0; k0 < XD; k0 += 4) {
    v2f a = *(const v2f*)(pa + k0);
    v2f b = *(const v2f*)(pb + k0);
    b.x *= bmsk;
    b.y *= bmsk;
    // 8 args: (neg_a, A, neg_b, B, c_mod, C, reuse_a, reuse_b)
    acc = __builtin_amdgcn_wmma_f32_16x16x4_f32(false, a, false, b,
                                                (short)0, acc, false, false);
  }

  const int   h     = h0 + mr;            // this lane's output column (N = lane%16)
  const float bias  = (h < HID && z == 0) ? b1[h] : 0.0f;
  const int   mbase = (lane < 16) ? 0 : 8;
  #pragma unroll
  for (int r = 0; r < 8; ++r) {
    const int n = n0 + mbase + r;         // M = r + (lane<16 ? 0 : 8)
    Dst[(size_t)n * HP + h] = (h < HID) ? (acc[r] + bias) : 0.0f;
  }
}

// ---------------------------------------------------------------------------
// Kernel 2: one block per row i. Computes T1[i,j] = softplus(b2 +
// sum_k relu(phx[j,k]+phy[i,k])*w2[k]) for all j, the diagonal term, and
// a deterministic in-block logsumexp over j. ReLU-inside-dot is not
// bilinear -> VALU fma path with hy_i / w2 staged in LDS, float4 vectorized.
// ---------------------------------------------------------------------------
__device__ __forceinline__ float softplus_f(float x) {
  return (x > 20.0f) ? x : log1pf(expf(x));
}

__global__ void pairwise_lse(const float* __restrict__ phx,
                             const float* __restrict__ phy,
                             const float* __restrict__ W2,
                             const float* __restrict__ b2,
                             float* __restrict__ lse,
                             float* __restrict__ diag) {
  __shared__ __align__(16) float hy_s[HP];
  __shared__ __align__(16) float w2_s[HP];
  __shared__ float red[256];

  const int i = blockIdx.x;
  const int t = threadIdx.x;              // 256 threads = 8 waves

  for (int k = t; k < HP; k += 256) {
    hy_s[k] = phy[(size_t)i * HP + k];
    w2_s[k] = (k < HID) ? W2[k] : 0.0f;   // zero pad kills padded-k contributions
  }
  __syncthreads();

  const float b2v = b2[0];

  float T1v[2];
  #pragma unroll
  for (int jj = 0; jj < 2; ++jj) {
    const int j = t + jj * 256;
    const float4* row = (const float4*)(phx + (size_t)j * HP);
    float s = 0.0f;
    #pragma unroll 4
    for (int k4 = 0; k4 < HP / 4; ++k4) {
      const float4 px = row[k4];
      const float4 hv = ((const float4*)hy_s)[k4];
      const float4 wv = ((const float4*)w2_s)[k4];
      s += fmaxf(px.x + hv.x, 0.0f) * wv.x;
      s += fmaxf(px.y + hv.y, 0.0f) * wv.y;
      s += fmaxf(px.z + hv.z, 0.0f) * wv.z;
      s += fmaxf(px.w + hv.w, 0.0f) * wv.w;
    }
    const float T1 = softplus_f(s + b2v);
    T1v[jj] = T1;
    if (j == i) diag[i] = T1;             // T0 is the diagonal of T1
  }

  // deterministic block reduction: max, then sum of exp(T1 - max)
  red[t] = fmaxf(T1v[0], T1v[1]);
  __syncthreads();
  for (int off = 128; off > 0; off >>= 1) {
    if (t < off) red[t] = fmaxf(red[t], red[t + off]);
    __syncthreads();
  }
  const float M = red[0];
  __syncthreads();

  red[t] = expf(T1v[0] - M) + expf(T1v[1] - M);
  __syncthreads();
  for (int off = 128; off > 0; off >>= 1) {
    if (t < off) red[t] += red[t + off];
    __syncthreads();
  }
  if (t == 0) lse[i] = M + logf(red[0]);
}

// ---------------------------------------------------------------------------
// Kernel 3: out = mean(diag) - (mean(lse) - log(N))   (deterministic tree)
// ---------------------------------------------------------------------------
__global__ void finalize(const float* __restrict__ lse,
                         const float* __restrict__ diag,
                         float* __restrict__ out) {
  __shared__ float r1[NS];
  __shared__ float r2[NS];
  const int t = threadIdx.x;              // NS threads
  r1[t] = diag[t];
  r2[t] = lse[t];
  __syncthreads();
  for (int off = NS / 2; off > 0; off >>= 1) {
    if (t < off) { r1[t] += r1[t + off]; r2[t] += r2[t + off]; }
    __syncthreads();
  }
  if (t == 0) {
    const float inv = 1.0f / (float)NS;
    out[0] = r1[0] * inv - (r2[0] * inv - logf((float)NS));
  }
}

extern "C" void kernel_launch(void* const* d_in, const int* in_sizes, int n_in,
                              void* d_out, int out_size, void* d_ws, size_t ws_size,
                              hipStream_t stream) {
  (void)in_sizes; (void)n_in; (void)out_size; (void)ws_size;
  const float* X  = (const float*)d_in[0];  // (512, 768)
  const float* Y  = (const float*)d_in[1];  // (512, 768)
  const float* W1 = (const float*)d_in[2];  // (300, 1536)
  const float* b1 = (const float*)d_in[3];  // (300,)
  const float* W2 = (const float*)d_in[4];  // (1, 300)
  const float* b2 = (const float*)d_in[5];  // (1,)

  float* ws   = (float*)d_ws;
  float* phx  = ws;                         // 512*304
  float* phy  = phx + (size_t)NS * HP;      // 512*304
  float* lse  = phy + (size_t)NS * HP;      // 512
  float* diag = lse + NS;                   // 512  (total ~1.25 MB)

  hid_gemm_wmma<<<dim3(NS / 16, HP / 16, 2), 32, 0, stream>>>(X, Y, W1, b1, phx, phy);
  pairwise_lse<<<NS, 256, 0, stream>>>(phx, phy, W2, b2, lse, diag);
  finalize<<<1, NS, 0, stream>>>(lse, diag, (float*)d_out);
}